// BasicBlockApprox_83683142795459
// MI455X (gfx1250) — compile-verified
//
#include <hip/hip_runtime.h>
#include <stdint.h>

// ---------------------------------------------------------------------------
// Problem constants (match the JAX reference)
// ---------------------------------------------------------------------------
#define BB     4
#define CIN    64
#define COUT   64
#define HH     32
#define WW     32
#define KK     3
#define BN_EPS 1e-5f

// tile config for kernel 1
#define OCG    16          // output channels per block
#define RT     4           // output rows per block
#define CCHUNK 16          // input channels staged per LDS pass
#define XROWS  (RT + 2)    // 6 staged rows (halo)
#define XPITCH 40          // padded row pitch in floats: w=-1 -> col 3, w=0 -> col 4 (16B aligned)

// ---------------------------------------------------------------------------
// gfx1250 feature probes (fall back cleanly on host pass / older toolchains)
// ---------------------------------------------------------------------------
#if defined(__gfx1250__) && __has_builtin(__builtin_amdgcn_global_load_async_to_lds_b128)
#define ASYNC_OK 1
#else
#define ASYNC_OK 0
#endif

#if defined(__gfx1250__) && __has_builtin(__builtin_amdgcn_wmma_f32_16x16x4_f32)
#define WMMA4_OK 1
#else
#define WMMA4_OK 0
#endif

typedef float v2f __attribute__((ext_vector_type(2)));
typedef float v8f __attribute__((ext_vector_type(8)));
typedef int   v4i __attribute__((vector_size(16)));   // matches builtin's expected element type

// 16-byte global -> LDS copy (async-to-LDS on gfx1250, plain copy otherwise)
__device__ __forceinline__ void g2l16(const float* g, float* l) {
#if ASYNC_OK
  typedef __attribute__((address_space(1))) v4i gv4i;   // global
  typedef __attribute__((address_space(3))) v4i lv4i;   // LDS
  gv4i* gp = (gv4i*)(uintptr_t)g;                       // flat addr == AS1 addr for global
  lv4i* lp = (lv4i*)(uint32_t)(uintptr_t)l;             // low 32 bits == LDS byte offset
  __builtin_amdgcn_global_load_async_to_lds_b128(gp, lp, 0, 0);
#else
  *(float4*)l = *(const float4*)g;
#endif
}

__device__ __forceinline__ void wait_async_copies() {
#if ASYNC_OK
#if __has_builtin(__builtin_amdgcn_s_wait_asynccnt)
  __builtin_amdgcn_s_wait_asynccnt(0);
#else
  asm volatile("s_wait_asynccnt 0x0" ::: "memory");
#endif
#endif
}

// ---------------------------------------------------------------------------
// Kernel 1: adder2d  out[b,o,h,w] = -sum_{c,kh,kw} |x[b,c,h+kh-1,w+kw-1] - W[o,c,kh,kw]|
// grid = (8 row-tiles, 4 oc-groups, 4 batch), block = 256 (8 waves)
// Raw (pre-BN) result written straight to d_out; kernel 3 normalizes in place.
// ---------------------------------------------------------------------------
__global__ __launch_bounds__(256) void k1_adder2d(const float* __restrict__ x,
                                                  const float* __restrict__ weight,
                                                  float* __restrict__ out) {
  __shared__ float xs[CCHUNK][XROWS][XPITCH];   // 15,360 B
  __shared__ float wls[OCG * CIN * KK * KK];    // 36,864 B

  const int tid = threadIdx.x;
  const int r0  = blockIdx.x * RT;              // first output row of tile
  const int ocg = blockIdx.y;                   // output-channel group (16 ch)
  const int b   = blockIdx.z;

  // ---- stage this group's weights (16 oc x 64 cin x 9) as float4s ----
  {
    const float4* wsrc = (const float4*)(weight + ocg * (OCG * CIN * KK * KK));
    float4* wdst = (float4*)wls;
#pragma unroll
    for (int i = 0; i < (OCG * CIN * KK * KK) / 4 / 256; ++i)
      wdst[tid + i * 256] = wsrc[tid + i * 256];
  }

  // ---- per-thread output mapping: 8 consecutive w positions ----
  const int ocl   = tid >> 4;                   // 0..15
  const int rem   = tid & 15;
  const int h_l   = rem >> 2;                   // 0..3
  const int wbase = (rem & 3) * 8;              // 0,8,16,24

  float acc[8];
#pragma unroll
  for (int i = 0; i < 8; ++i) acc[i] = 0.0f;

  for (int chunk = 0; chunk < CIN / CCHUNK; ++chunk) {
    const int cbase = chunk * CCHUNK;
    __syncthreads();                            // previous chunk fully consumed

    // zero halo columns + any out-of-image rows
    if (tid < CCHUNK * XROWS) {
      const int c = tid / XROWS, row = tid % XROWS;
      const int gr = r0 - 1 + row;
      xs[c][row][3]  = 0.0f;                    // w = -1
      xs[c][row][36] = 0.0f;                    // w = 32
      if (gr < 0 || gr > HH - 1) {
#pragma unroll
        for (int j = 4; j < 36; ++j) xs[c][row][j] = 0.0f;
      }
    }
    __syncthreads();

    // async-stage valid rows: 16 ch x 6 rows x 8 float4
    for (int idx = tid; idx < CCHUNK * XROWS * 8; idx += 256) {
      const int c = idx / (XROWS * 8);
      const int r2 = idx % (XROWS * 8);
      const int row = r2 >> 3, q = r2 & 7;
      const int gr = r0 - 1 + row;
      if (gr >= 0 && gr < HH)
        g2l16(x + ((((b * CIN) + cbase + c) * HH + gr) << 5) + q * 4,
              &xs[c][row][4 + q * 4]);
    }
    wait_async_copies();
    __syncthreads();

    // accumulate |x - w| over this channel chunk
    for (int cc = 0; cc < CCHUNK; ++cc) {
      const float* wrow = &wls[(ocl * CIN + cbase + cc) * 9];
#pragma unroll
      for (int kh = 0; kh < KK; ++kh) {
        const float* xrow = &xs[cc][h_l + kh][wbase];
        float xr[16];
#pragma unroll
        for (int q = 0; q < 4; ++q)
          *(float4*)&xr[q * 4] = *(const float4*)&xrow[q * 4];
#pragma unroll
        for (int kw = 0; kw < KK; ++kw) {
          const float wv = wrow[kh * 3 + kw];
#pragma unroll
          for (int i = 0; i < 8; ++i)
            acc[i] += __builtin_fabsf(xr[i + kw + 3] - wv);
        }
      }
    }
  }

  // store raw adder output (negated)
  const int oc = ocg * OCG + ocl;
  float* po = out + (((b * COUT) + oc) << 10) + ((r0 + h_l) << 5) + wbase;
#pragma unroll
  for (int q = 0; q < 2; ++q) {
    float4 st;
    st.x = -acc[q * 4 + 0]; st.y = -acc[q * 4 + 1];
    st.z = -acc[q * 4 + 2]; st.w = -acc[q * 4 + 3];
    *(float4*)(po + q * 4) = st;
  }
}

// ---------------------------------------------------------------------------
// Kernel 2: BN batch statistics via WMMA row-reduction.
// D = A(16x4) x ones(4x16) + C accumulates per-channel sums into every column.
// One block per 16-channel group; 8 waves each reduce 512 of the 4096 samples.
// ---------------------------------------------------------------------------
__global__ __launch_bounds__(256) void k2_bn_stats(const float* __restrict__ raw,
                                                   float* __restrict__ meanArr,
                                                   float* __restrict__ rstdArr) {
  const int ocbase = blockIdx.x * 16;
  const int tid  = threadIdx.x;
  const int lane = tid & 31;
  const int wave = tid >> 5;
  const int m    = lane & 15;                   // channel within group
  const int k0   = (lane >> 4) << 1;            // K offset 0 or 2
  const int ch   = ocbase + m;

  __shared__ float s_sum[8][16];
  __shared__ float s_sq[8][16];

#if WMMA4_OK
  v8f csum = {};
  v8f csq  = {};
  v2f ones; ones.x = 1.0f; ones.y = 1.0f;
#else
  float psum = 0.0f, psq = 0.0f;
#endif

#pragma unroll 4
  for (int it = 0; it < 128; ++it) {
    const int j   = wave * 512 + it * 4 + k0;   // sample index 0..4095
    const int bim = j >> 10;
    const int pix = j & 1023;
    const float2 v = *(const float2*)(raw + (((bim * COUT) + ch) << 10) + pix);
#if WMMA4_OK
    v2f a;  a.x  = v.x;       a.y  = v.y;
    v2f a2; a2.x = v.x * v.x; a2.y = v.y * v.y;
    csum = __builtin_amdgcn_wmma_f32_16x16x4_f32(false, a,  false, ones, (short)0, csum, false, false);
    csq  = __builtin_amdgcn_wmma_f32_16x16x4_f32(false, a2, false, ones, (short)0, csq,  false, false);
#else
    psum += v.x + v.y;
    psq  += v.x * v.x + v.y * v.y;
#endif
  }

#if WMMA4_OK
  // C/D layout: VGPR r, lanes 0-15 -> M=r; lanes 16-31 -> M=r+8 (column N identical)
  if (lane == 0 || lane == 16) {
    const int cb = (lane >> 4) * 8;
#pragma unroll
    for (int r = 0; r < 8; ++r) {
      s_sum[wave][cb + r] = csum[r];
      s_sq [wave][cb + r] = csq[r];
    }
  }
#else
  psum += __shfl_xor(psum, 16, 32);
  psq  += __shfl_xor(psq, 16, 32);
  if (lane < 16) { s_sum[wave][lane] = psum; s_sq[wave][lane] = psq; }
#endif
  __syncthreads();

  if (tid < 16) {
    float s = 0.0f, q = 0.0f;
#pragma unroll
    for (int w2 = 0; w2 < 8; ++w2) { s += s_sum[w2][tid]; q += s_sq[w2][tid]; }
    const float inv_n = 1.0f / (float)(BB * HH * WW);
    const float mean  = s * inv_n;
    const float var   = q * inv_n - mean * mean;
    meanArr[ocbase + tid] = mean;
    rstdArr[ocbase + tid] = rsqrtf(var + BN_EPS);
  }
}

// ---------------------------------------------------------------------------
// Kernel 3: in-place BN affine + ReLU over d_out
// ---------------------------------------------------------------------------
__global__ __launch_bounds__(256) void k3_bn_relu(float* __restrict__ out,
                                                  const float* __restrict__ meanArr,
                                                  const float* __restrict__ rstdArr,
                                                  const float* __restrict__ gamma,
                                                  const float* __restrict__ beta) {
  const int idx4 = blockIdx.x * blockDim.x + threadIdx.x;   // 0..65535
  const int base = idx4 << 2;
  const int ch   = (base >> 10) & (COUT - 1);               // 4 elems share a channel
  const float mu = meanArr[ch];
  const float sc = rstdArr[ch] * gamma[ch];
  const float bt = beta[ch];
  float4 v = *(float4*)(out + base);
  v.x = fmaxf((v.x - mu) * sc + bt, 0.0f);
  v.y = fmaxf((v.y - mu) * sc + bt, 0.0f);
  v.z = fmaxf((v.z - mu) * sc + bt, 0.0f);
  v.w = fmaxf((v.w - mu) * sc + bt, 0.0f);
  *(float4*)(out + base) = v;
}

// ---------------------------------------------------------------------------
extern "C" void kernel_launch(void* const* d_in, const int* in_sizes, int n_in,
                              void* d_out, int out_size, void* d_ws, size_t ws_size,
                              hipStream_t stream) {
  const float* x      = (const float*)d_in[0];
  const float* weight = (const float*)d_in[1];
  const float* gamma  = (const float*)d_in[2];
  const float* beta   = (const float*)d_in[3];
  float* out = (float*)d_out;

  float* meanArr = (float*)d_ws;        // 64 floats
  float* rstdArr = meanArr + COUT;      // 64 floats

  dim3 g1(HH / RT, COUT / OCG, BB);     // 8 x 4 x 4 = 128 blocks
  k1_adder2d<<<g1, 256, 0, stream>>>(x, weight, out);
  k2_bn_stats<<<COUT / 16, 256, 0, stream>>>(out, meanArr, rstdArr);
  k3_bn_relu<<<(BB * COUT * HH * WW) / 4 / 256, 256, 0, stream>>>(out, meanArr, rstdArr, gamma, beta);
}